// Net_75505525064500
// MI455X (gfx1250) — compile-verified
//
#include <hip/hip_runtime.h>
#include <cfloat>
#include <climits>
#include <cstdint>

typedef __attribute__((ext_vector_type(2))) float v2f;
typedef __attribute__((ext_vector_type(8))) float v8f;

#define NUMK 27

__device__ __forceinline__ float eluf(float x) {
    return x > 0.0f ? x : (expf(x) - 1.0f);
}

// Cooperative async Global->LDS copy (16B granules), CDNA5 async-VMEM path.
// LDS byte address is taken from the vdst VGPR; flat addr low 32 bits of a __shared__ pointer
// are the group-segment byte offset, which is exactly what VDST expects.
// nfloats must be a multiple of 4. Follow with __syncthreads() after return.
__device__ __forceinline__ void async_copy_to_lds(const float* __restrict__ gsrc,
                                                  float* lds_dst, int nfloats) {
    int nchunks = nfloats >> 2;
    for (int chunk = threadIdx.x; chunk < nchunks; chunk += blockDim.x) {
        unsigned lds_addr =
            (unsigned)(reinterpret_cast<uintptr_t>(lds_dst + chunk * 4));
        unsigned long long gaddr =
            (unsigned long long)(uintptr_t)(gsrc + chunk * 4);
        asm volatile("global_load_async_to_lds_b128 %0, %1, off"
                     :: "v"(lds_addr), "v"(gaddr)
                     : "memory");
    }
    asm volatile("s_wait_asynccnt 0x0" ::: "memory");
}

// Linear open B-spline basis (degree 1, K=3 per dim), matches reference.
__device__ __forceinline__ void bspline_basis(float a0, float a1, float a2,
                                              float wgt[8], int kidx[8]) {
    float a[3] = {a0, a1, a2};
    float f[3];
    int i0[3];
#pragma unroll
    for (int d = 0; d < 3; ++d) {
        float v = a[d] * 2.0f;                  // (K-1) = 2
        float fl = floorf(v);
        fl = fminf(fmaxf(fl, 0.0f), 1.0f);      // clip to [0, K-2]
        f[d] = v - fl;
        i0[d] = (int)fl;
    }
    const int strides[3] = {1, 3, 9};
#pragma unroll
    for (int s = 0; s < 8; ++s) {
        float w = 1.0f;
        int k = 0;
#pragma unroll
        for (int d = 0; d < 3; ++d) {
            int bit = (s >> d) & 1;
            w *= bit ? f[d] : (1.0f - f[d]);
            k += (i0[d] + bit) * strides[d];
        }
        wgt[s] = w;
        kidx[s] = k;
    }
}

// Float atomic max via signed/unsigned int ordering trick.
__device__ __forceinline__ void atomicMaxF(float* addr, float val) {
    if (val >= 0.0f) atomicMax((int*)addr, __float_as_int(val));
    else             atomicMin((unsigned int*)addr, __float_as_uint(val));
}

__global__ void fill_f32(float* __restrict__ p, float v, int n) {
    int i = blockIdx.x * blockDim.x + threadIdx.x;
    if (i < n) p[i] = v;
}

__global__ void fill_i32(int* __restrict__ p, int v, int n) {
    int i = blockIdx.x * blockDim.x + threadIdx.x;
    if (i < n) p[i] = v;
}

// Level-1 edge kernel: msg[e] = sum_c basis[e,c] * (x[src] @ W1[kidx[e,c]])
__global__ __launch_bounds__(256) void edge1_kernel(
    const int* __restrict__ src, const int* __restrict__ dst,
    const float* __restrict__ attr, const float* __restrict__ x,
    const float* __restrict__ W1, float* __restrict__ agg1,
    float* __restrict__ cnt1, int E) {
    __shared__ float sW[NUMK * 16];   // [27][2][8]
    async_copy_to_lds(W1, sW, NUMK * 16);
    __syncthreads();
    int e = blockIdx.x * blockDim.x + threadIdx.x;
    if (e >= E) return;
    int s = src[e], d = dst[e];
    float wgt[8]; int kidx[8];
    bspline_basis(attr[e * 3 + 0], attr[e * 3 + 1], attr[e * 3 + 2], wgt, kidx);
    float x0 = x[s * 2 + 0], x1v = x[s * 2 + 1];
    float msg[8] = {0.f, 0.f, 0.f, 0.f, 0.f, 0.f, 0.f, 0.f};
#pragma unroll
    for (int c = 0; c < 8; ++c) {
        const float* w = &sW[kidx[c] * 16];
        float b0 = wgt[c] * x0, b1 = wgt[c] * x1v;
#pragma unroll
        for (int o = 0; o < 8; ++o) msg[o] += b0 * w[o] + b1 * w[8 + o];
    }
#pragma unroll
    for (int o = 0; o < 8; ++o) atomicAdd(&agg1[d * 8 + o], msg[o]);
    atomicAdd(&cnt1[d], 1.0f);
}

// Level-1 node epilogue: mean + root weight + bias + ELU, then max-pool into x1p.
__global__ __launch_bounds__(256) void node1_kernel(
    const float* __restrict__ agg1, const float* __restrict__ cnt1,
    const float* __restrict__ x, const float* __restrict__ root1,
    const float* __restrict__ b1, const int* __restrict__ cluster1,
    const int* __restrict__ batch, float* __restrict__ x1p,
    int* __restrict__ bat1, int N) {
    int n = blockIdx.x * blockDim.x + threadIdx.x;
    if (n >= N) return;
    float inv = 1.0f / fmaxf(cnt1[n], 1.0f);
    float x0 = x[n * 2 + 0], x1v = x[n * 2 + 1];
    int cl = cluster1[n];
#pragma unroll
    for (int o = 0; o < 8; ++o) {
        float v = agg1[n * 8 + o] * inv + x0 * root1[o] + x1v * root1[8 + o] + b1[o];
        atomicMaxF(&x1p[cl * 8 + o], eluf(v));
    }
    atomicMax(&bat1[cl], batch[n]);
}

__global__ __launch_bounds__(256) void fixup1_kernel(
    float* __restrict__ x1p, int* __restrict__ bat1, int N1) {
    int n = blockIdx.x * blockDim.x + threadIdx.x;
    if (n >= N1) return;
#pragma unroll
    for (int o = 0; o < 8; ++o) {
        float v = x1p[n * 8 + o];
        if (v == -FLT_MAX) x1p[n * 8 + o] = 0.0f;   // empty cluster -> 0
    }
    bat1[n] = max(bat1[n], 0);
}

// Level-2 edge kernel: Cin=8, Cout=16, W2 (13.8 KB) resident in LDS,
// staged via async global->LDS DMA.
__global__ __launch_bounds__(256) void edge2_kernel(
    const int* __restrict__ src, const int* __restrict__ dst,
    const int* __restrict__ cluster1, const float* __restrict__ attr,
    const float* __restrict__ x1p, const float* __restrict__ W2,
    float* __restrict__ agg2, float* __restrict__ cnt2, int E) {
    __shared__ float sW[NUMK * 128];  // [27][8][16]
    async_copy_to_lds(W2, sW, NUMK * 128);
    __syncthreads();
    int e = blockIdx.x * blockDim.x + threadIdx.x;
    if (e >= E) return;
    int s2 = cluster1[src[e]], d2 = cluster1[dst[e]];
    float wgt[8]; int kidx[8];
    bspline_basis(attr[e * 3 + 0], attr[e * 3 + 1], attr[e * 3 + 2], wgt, kidx);
    float xv[8];
#pragma unroll
    for (int i = 0; i < 8; ++i) xv[i] = x1p[s2 * 8 + i];
    float msg[16];
#pragma unroll
    for (int o = 0; o < 16; ++o) msg[o] = 0.0f;
#pragma unroll
    for (int c = 0; c < 8; ++c) {
        const float* w = &sW[kidx[c] * 128];
#pragma unroll
        for (int i = 0; i < 8; ++i) {
            float xb = wgt[c] * xv[i];
#pragma unroll
            for (int o = 0; o < 16; ++o) msg[o] += xb * w[i * 16 + o];
        }
    }
#pragma unroll
    for (int o = 0; o < 16; ++o) atomicAdd(&agg2[d2 * 16 + o], msg[o]);
    atomicAdd(&cnt2[d2], 1.0f);
}

__global__ __launch_bounds__(256) void node2_kernel(
    const float* __restrict__ agg2, const float* __restrict__ cnt2,
    const float* __restrict__ x1p, const float* __restrict__ root2,
    const float* __restrict__ b2, const int* __restrict__ cluster2,
    const int* __restrict__ bat1, float* __restrict__ x2p,
    int* __restrict__ bat2, int N1) {
    int n = blockIdx.x * blockDim.x + threadIdx.x;
    if (n >= N1) return;
    float inv = 1.0f / fmaxf(cnt2[n], 1.0f);
    float xv[8];
#pragma unroll
    for (int i = 0; i < 8; ++i) xv[i] = x1p[n * 8 + i];
    int cl = cluster2[n];
#pragma unroll
    for (int o = 0; o < 16; ++o) {
        float v = agg2[n * 16 + o] * inv + b2[o];
#pragma unroll
        for (int i = 0; i < 8; ++i) v += xv[i] * root2[i * 16 + o];
        atomicMaxF(&x2p[cl * 16 + o], eluf(v));
    }
    atomicMax(&bat2[cl], bat1[n]);
}

// Fixup x2 + scatter-mean accumulation over graphs.
__global__ __launch_bounds__(256) void graph_accum_kernel(
    const float* __restrict__ x2p, const int* __restrict__ bat2,
    float* __restrict__ gsum, float* __restrict__ gcnt, int N2) {
    int n = blockIdx.x * blockDim.x + threadIdx.x;
    if (n >= N2) return;
    int b = max(bat2[n], 0);
#pragma unroll
    for (int o = 0; o < 16; ++o) {
        float v = x2p[n * 16 + o];
        if (v == -FLT_MAX) v = 0.0f;
        atomicAdd(&gsum[b * 16 + o], v);
    }
    atomicAdd(&gcnt[b], 1.0f);
}

// Final MLP on one wave32 using V_WMMA_F32_16X16X4_F32.
// g[16,16] @ fc1_w[16,64]: 4 N-tiles x 4 K-steps of 16x16x4 f32 WMMA.
__global__ __launch_bounds__(32) void mlp_kernel(
    const float* __restrict__ gsum, const float* __restrict__ gcnt,
    const float* __restrict__ fc1_w, const float* __restrict__ fc1_b,
    const float* __restrict__ fc2_w, const float* __restrict__ fc2_b,
    float* __restrict__ out) {
    __shared__ float g[16 * 16];
    __shared__ float h1[16 * 64];
    int lane = threadIdx.x;
#pragma unroll
    for (int i = 0; i < 8; ++i) {
        int idx = lane + 32 * i;
        int row = idx >> 4;
        g[idx] = gsum[idx] / fmaxf(gcnt[row], 1.0f);
    }
    __syncthreads();
    int mrow = lane & 15;
    int khalf = lane >> 4;   // 0: lanes 0-15 (K pair 0,1), 1: lanes 16-31 (K pair 2,3)
#pragma unroll
    for (int t = 0; t < 4; ++t) {
        v8f c = {0.f, 0.f, 0.f, 0.f, 0.f, 0.f, 0.f, 0.f};
#pragma unroll
        for (int kc = 0; kc < 4; ++kc) {
            int k0 = 4 * kc + 2 * khalf;
            v2f a, b;
            a.x = g[mrow * 16 + k0];                 // A: 16x4 chunk of g
            a.y = g[mrow * 16 + k0 + 1];
            b.x = fc1_w[k0 * 64 + 16 * t + mrow];    // B: 4x16 chunk of fc1_w
            b.y = fc1_w[(k0 + 1) * 64 + 16 * t + mrow];
            c = __builtin_amdgcn_wmma_f32_16x16x4_f32(
                false, a, false, b, (short)0, c, false, false);
        }
#pragma unroll
        for (int i = 0; i < 8; ++i) {
            int m = (lane < 16) ? i : (i + 8);       // C/D layout: half-wave M split
            int ncol = 16 * t + mrow;
            h1[m * 64 + ncol] = eluf(c[i] + fc1_b[ncol]);
        }
    }
    __syncthreads();
    if (lane < 16) {
        float acc = fc2_b[0];
#pragma unroll 8
        for (int j = 0; j < 64; ++j) acc += h1[lane * 64 + j] * fc2_w[j];
        out[lane] = eluf(acc);
    }
}

extern "C" void kernel_launch(void* const* d_in, const int* in_sizes, int n_in,
                              void* d_out, int out_size, void* d_ws, size_t ws_size,
                              hipStream_t stream) {
    const float* x         = (const float*)d_in[0];
    const int*   edge_idx  = (const int*)d_in[1];
    const float* edge_attr = (const float*)d_in[2];
    const int*   batch     = (const int*)d_in[3];
    const int*   cluster1  = (const int*)d_in[4];
    const int*   cluster2  = (const int*)d_in[5];
    const float* W1        = (const float*)d_in[6];
    const float* root1     = (const float*)d_in[7];
    const float* b1        = (const float*)d_in[8];
    const float* W2        = (const float*)d_in[9];
    const float* root2     = (const float*)d_in[10];
    const float* b2        = (const float*)d_in[11];
    const float* fc1_w     = (const float*)d_in[12];
    const float* fc1_b     = (const float*)d_in[13];
    const float* fc2_w     = (const float*)d_in[14];
    const float* fc2_b     = (const float*)d_in[15];

    const int N  = in_sizes[0] / 2;   // 80000
    const int E  = in_sizes[1] / 2;   // 1280000
    const int N1 = in_sizes[5];       // 40000 (cluster2 has N1 entries)
    const int N2 = N1 / 2;            // 20000
    const int B  = 16;

    const int* src = edge_idx;
    const int* dst = edge_idx + E;

    char* ws = (char*)d_ws;
    size_t off = 0;
    auto wsalloc = [&](size_t bytes) -> void* {
        void* p = ws + off;
        off += (bytes + 255) & ~(size_t)255;
        return p;
    };
    float* agg1 = (float*)wsalloc((size_t)N * 8 * 4);
    float* cnt1 = (float*)wsalloc((size_t)N * 4);
    float* x1p  = (float*)wsalloc((size_t)N1 * 8 * 4);
    int*   bat1 = (int*)wsalloc((size_t)N1 * 4);
    float* agg2 = (float*)wsalloc((size_t)N1 * 16 * 4);
    float* cnt2 = (float*)wsalloc((size_t)N1 * 4);
    float* x2p  = (float*)wsalloc((size_t)N2 * 16 * 4);
    int*   bat2 = (int*)wsalloc((size_t)N2 * 4);
    float* gsum = (float*)wsalloc((size_t)B * 16 * 4);
    float* gcnt = (float*)wsalloc((size_t)B * 4);
    (void)ws_size; (void)n_in; (void)out_size;

    auto cdiv = [](int a, int b) { return (a + b - 1) / b; };

    fill_f32<<<cdiv(N * 8, 256), 256, 0, stream>>>(agg1, 0.0f, N * 8);
    fill_f32<<<cdiv(N, 256), 256, 0, stream>>>(cnt1, 0.0f, N);
    fill_f32<<<cdiv(N1 * 8, 256), 256, 0, stream>>>(x1p, -FLT_MAX, N1 * 8);
    fill_i32<<<cdiv(N1, 256), 256, 0, stream>>>(bat1, INT_MIN, N1);
    fill_f32<<<cdiv(N1 * 16, 256), 256, 0, stream>>>(agg2, 0.0f, N1 * 16);
    fill_f32<<<cdiv(N1, 256), 256, 0, stream>>>(cnt2, 0.0f, N1);
    fill_f32<<<cdiv(N2 * 16, 256), 256, 0, stream>>>(x2p, -FLT_MAX, N2 * 16);
    fill_i32<<<cdiv(N2, 256), 256, 0, stream>>>(bat2, INT_MIN, N2);
    fill_f32<<<cdiv(B * 16, 256), 256, 0, stream>>>(gsum, 0.0f, B * 16);
    fill_f32<<<1, 256, 0, stream>>>(gcnt, 0.0f, B);

    edge1_kernel<<<cdiv(E, 256), 256, 0, stream>>>(src, dst, edge_attr, x, W1,
                                                   agg1, cnt1, E);
    node1_kernel<<<cdiv(N, 256), 256, 0, stream>>>(agg1, cnt1, x, root1, b1,
                                                   cluster1, batch, x1p, bat1, N);
    fixup1_kernel<<<cdiv(N1, 256), 256, 0, stream>>>(x1p, bat1, N1);
    edge2_kernel<<<cdiv(E, 256), 256, 0, stream>>>(src, dst, cluster1, edge_attr,
                                                   x1p, W2, agg2, cnt2, E);
    node2_kernel<<<cdiv(N1, 256), 256, 0, stream>>>(agg2, cnt2, x1p, root2, b2,
                                                    cluster2, bat1, x2p, bat2, N1);
    graph_accum_kernel<<<cdiv(N2, 256), 256, 0, stream>>>(x2p, bat2, gsum, gcnt, N2);
    mlp_kernel<<<1, 32, 0, stream>>>(gsum, gcnt, fc1_w, fc1_b, fc2_w, fc2_b,
                                     (float*)d_out);
}